// SpringMassSystem_43903155699878
// MI455X (gfx1250) — compile-verified
//
#include <hip/hip_runtime.h>
#include <hip/hip_bf16.h>

#define N_SUBSTEPS 20
#define DT_F       5e-05f
#define DASHPOT_F  100.0f
#define DRAG_F     3.0f
#define Y_MIN_F    1000.0f
#define Y_MAX_F    100000.0f
#define GRAV_Z     (-9.8f)

struct __align__(16) SpringRec {
    int   i1;
    int   i2;
    float rest;
    float coeff;   // mask * min(exp(Y), Y_MAX); coeff > 0 <=> mask == 1
};

// ---------------------------------------------------------------------------
// Repack tight (nvert,3) x/v into 16B-aligned float4 state (w unused) and
// zero the float4 force accumulator. One thread per vertex.
// ---------------------------------------------------------------------------
__global__ __launch_bounds__(256) void init_state_kernel(
    const float* __restrict__ x_in, const float* __restrict__ v_in,
    float4* __restrict__ xb, float4* __restrict__ vb, float4* __restrict__ f,
    int nvert)
{
    int i = blockIdx.x * blockDim.x + threadIdx.x;
    if (i >= nvert) return;
    xb[i] = make_float4(x_in[3 * i], x_in[3 * i + 1], x_in[3 * i + 2], 0.0f);
    vb[i] = make_float4(v_in[3 * i], v_in[3 * i + 1], v_in[3 * i + 2], 0.0f);
    f[i]  = make_float4(0.0f, 0.0f, 0.0f, 0.0f);
}

// ---------------------------------------------------------------------------
// Pack per-spring static data into one 16B record; fold exp/mask/min once
// per call (it is loop-invariant across all 20 substeps).
// ---------------------------------------------------------------------------
__global__ __launch_bounds__(256) void build_rec_kernel(
    const int*   __restrict__ springs,       // (nspr, 2)
    const float* __restrict__ rest_lengths,
    const float* __restrict__ spring_Y,
    SpringRec*   __restrict__ rec,
    int nspr)
{
    int i = blockIdx.x * blockDim.x + threadIdx.x;
    if (i >= nspr) return;
    float sY    = expf(spring_Y[i]);
    float coeff = (sY > Y_MIN_F) ? fminf(sY, Y_MAX_F) : 0.0f;
    SpringRec r;
    r.i1    = springs[2 * i + 0];
    r.i2    = springs[2 * i + 1];
    r.rest  = rest_lengths[i];
    r.coeff = coeff;
    rec[i] = r;
}

// ---------------------------------------------------------------------------
// Spring force kernel. Streams SpringRec via CDNA5 async global->LDS copies
// (per-wave private double buffer, no barriers), gathers x/v as single
// aligned b128 loads, scatter-adds forces with global f32 atomics (L2-
// resident). Emits sf (elastic part only, non-temporal) on the last substep.
// ---------------------------------------------------------------------------
#define SPR_TPB 256

__global__ __launch_bounds__(SPR_TPB) void spring_kernel(
    const SpringRec* __restrict__ rec,
    const float4*    __restrict__ x,
    const float4*    __restrict__ v,
    float*           __restrict__ f,       // float4-strided: f[4*i + c]
    float*           __restrict__ sf_out,
    int nspr, int write_sf)
{
    __shared__ SpringRec s_stage[2][SPR_TPB];   // 8 KB; each lane owns its slot

    const int tid    = threadIdx.x;
    const int gid    = blockIdx.x * SPR_TPB + tid;
    const int stride = gridDim.x * SPR_TPB;
    const int iters  = (nspr + stride - 1) / stride;   // uniform across grid

    // LDS byte offsets of this lane's two staging slots (addrspace(3) offset
    // == low 32 bits of the generic pointer).
    const unsigned lds0 = (unsigned)(size_t)&s_stage[0][tid];
    const unsigned lds1 = (unsigned)(size_t)&s_stage[1][tid];

    // Prefetch tile 0 (clamped so every lane issues a valid address).
    {
        int i0 = gid; if (i0 >= nspr) i0 = nspr - 1;
        unsigned voff = (unsigned)i0 * 16u;
        asm volatile("global_load_async_to_lds_b128 %0, %1, %2"
                     :: "v"(lds0), "v"(voff), "s"(rec) : "memory");
    }

    for (int it = 0; it < iters; ++it) {
        const int      buf    = it & 1;
        const unsigned ldsNxt = (buf ? lds0 : lds1);

        if (it + 1 < iters) {
            int inx = gid + (it + 1) * stride;
            if (inx >= nspr) inx = nspr - 1;
            unsigned voff = (unsigned)inx * 16u;
            asm volatile("global_load_async_to_lds_b128 %0, %1, %2"
                         :: "v"(ldsNxt), "v"(voff), "s"(rec) : "memory");
            // allow the just-issued copy to stay in flight; wait for current
            asm volatile("s_wait_asynccnt 0x1" ::: "memory");
        } else {
            asm volatile("s_wait_asynccnt 0x0" ::: "memory");
        }

        const SpringRec r = s_stage[buf][tid];     // ds_load_b128

        const int spr = gid + it * stride;
        if (spr < nspr) {
            const float4 p1 = x[r.i1];             // single aligned b128 gather
            const float4 p2 = x[r.i2];
            const float dx = p2.x - p1.x;
            const float dy = p2.y - p1.y;
            const float dz = p2.z - p1.z;
            const float nrm  = sqrtf(dx * dx + dy * dy + dz * dz);
            const float inv  = 1.0f / fmaxf(nrm, 1e-06f);
            const float ddx = dx * inv, ddy = dy * inv, ddz = dz * inv;

            const float sfs = r.coeff * (nrm / r.rest - 1.0f);
            const float sfx = sfs * ddx;
            const float sfy = sfs * ddy;
            const float sfz = sfs * ddz;

            if (write_sf) {
                __builtin_nontemporal_store(sfx, &sf_out[3 * spr + 0]);
                __builtin_nontemporal_store(sfy, &sf_out[3 * spr + 1]);
                __builtin_nontemporal_store(sfz, &sf_out[3 * spr + 2]);
            }

            const float4 q1 = v[r.i1];
            const float4 q2 = v[r.i2];
            const float vrel = (q2.x - q1.x) * ddx + (q2.y - q1.y) * ddy +
                               (q2.z - q1.z) * ddz;
            const float dmp = (r.coeff > 0.0f) ? (DASHPOT_F * vrel) : 0.0f;

            const float tx = sfx + dmp * ddx;
            const float ty = sfy + dmp * ddy;
            const float tz = sfz + dmp * ddz;

            atomicAdd(&f[4 * r.i1 + 0],  tx);
            atomicAdd(&f[4 * r.i1 + 1],  ty);
            atomicAdd(&f[4 * r.i1 + 2],  tz);
            atomicAdd(&f[4 * r.i2 + 0], -tx);
            atomicAdd(&f[4 * r.i2 + 1], -ty);
            atomicAdd(&f[4 * r.i2 + 2], -tz);
        }
    }
}

// ---------------------------------------------------------------------------
// Vertex integration: v' = (v + DT*g + DT*f/m)*drag ; x' = x + DT*v'.
// All state is float4 (single b128 load/store per array). Re-zeroes f for the
// next substep. Writes final tight-layout x to d_out on the last substep.
// ---------------------------------------------------------------------------
__global__ __launch_bounds__(256) void vertex_kernel(
    float4* __restrict__ x, float4* __restrict__ v, float4* __restrict__ f,
    const float* __restrict__ masses,
    float* __restrict__ x_out,
    int nvert, int write_x)
{
    int i = blockIdx.x * blockDim.x + threadIdx.x;
    if (i >= nvert) return;

    const float drag = expf(-DT_F * DRAG_F);  // constant-folded
    const float invm = DT_F / masses[i];

    const float4 fi = f[i];
    f[i] = make_float4(0.0f, 0.0f, 0.0f, 0.0f);

    const float4 vi = v[i];
    const float vx = (vi.x + fi.x * invm) * drag;
    const float vy = (vi.y + fi.y * invm) * drag;
    const float vz = (vi.z + fi.z * invm + DT_F * GRAV_Z) * drag;

    const float4 xi = x[i];
    const float nx = xi.x + DT_F * vx;
    const float ny = xi.y + DT_F * vy;
    const float nz = xi.z + DT_F * vz;

    v[i] = make_float4(vx, vy, vz, 0.0f);
    x[i] = make_float4(nx, ny, nz, 0.0f);

    if (write_x) {
        __builtin_nontemporal_store(nx, &x_out[3 * i + 0]);
        __builtin_nontemporal_store(ny, &x_out[3 * i + 1]);
        __builtin_nontemporal_store(nz, &x_out[3 * i + 2]);
    }
}

// ---------------------------------------------------------------------------
// Host-side launch. Inputs: x, v, rest_lengths, masses, spring_Y, springs.
// d_out = [x_final (3*nvert) | sf_last (3*nspr)].
// ---------------------------------------------------------------------------
extern "C" void kernel_launch(void* const* d_in, const int* in_sizes, int n_in,
                              void* d_out, int out_size, void* d_ws, size_t ws_size,
                              hipStream_t stream)
{
    const float* x_in   = (const float*)d_in[0];
    const float* v_in   = (const float*)d_in[1];
    const float* rest   = (const float*)d_in[2];
    const float* masses = (const float*)d_in[3];
    const float* sprY   = (const float*)d_in[4];
    const int*   sprs   = (const int*)d_in[5];

    const int nvert = in_sizes[0] / 3;
    const int nspr  = in_sizes[2];

    // Workspace layout (all 16B-aligned):
    float4*    f   = (float4*)d_ws;                 // nvert float4
    float4*    xb  = f  + (size_t)nvert;            // nvert float4
    float4*    vb  = xb + (size_t)nvert;            // nvert float4
    SpringRec* rec = (SpringRec*)(vb + (size_t)nvert); // nspr * 16B

    float* out_x  = (float*)d_out;
    float* out_sf = out_x + 3 * (size_t)nvert;

    init_state_kernel<<<(nvert + 255) / 256, 256, 0, stream>>>(x_in, v_in, xb, vb, f, nvert);
    build_rec_kernel<<<(nspr + 255) / 256, 256, 0, stream>>>(sprs, rest, sprY, rec, nspr);

    const int sprBlocks = 1024;               // 262144 threads -> ~7 tiles/thread
    const int vtxBlocks = (nvert + 255) / 256;

    for (int s = 0; s < N_SUBSTEPS; ++s) {
        const int last = (s == N_SUBSTEPS - 1) ? 1 : 0;
        spring_kernel<<<sprBlocks, SPR_TPB, 0, stream>>>(
            rec, xb, vb, (float*)f, out_sf, nspr, last);
        vertex_kernel<<<vtxBlocks, 256, 0, stream>>>(
            xb, vb, f, masses, out_x, nvert, last);
    }
}